// MMDLoss_16939351015447
// MI455X (gfx1250) — compile-verified
//
#include <hip/hip_runtime.h>

typedef __attribute__((ext_vector_type(2))) float v2f;
typedef __attribute__((ext_vector_type(8))) float v8f;

#define NS      4096   // rows in source (and target)
#define NTOT    8192   // NS * 2
#define D       512    // feature dim
#define TS      64     // macro tile edge per wave (4x4 WMMA tiles)
#define TPD     (NTOT / TS)          // 128 macro tiles per dimension
#define NTILES  (TPD * TPD)          // 16384 wave-tiles
#define WAVES_PER_BLOCK 8

// workspace layout (floats)
#define WS_SQ      0          // [8192] row squared norms
#define WS_COLSUM  8192       // [512]  per-dim column sums
#define WS_BW      8704       // [1]    bandwidth
#define WS_ACC     8705       // [3]    S_xx, S_yy, S_xy(both off-diag blocks)

__device__ __forceinline__ const float* row_ptr(const float* __restrict__ src,
                                                const float* __restrict__ tgt,
                                                int row) {
    return (row < NS) ? (src + (size_t)row * D) : (tgt + (size_t)(row - NS) * D);
}

// ---- zero the reduction slots (colsum + bw + accumulators) -----------------
__global__ void k_init(float* __restrict__ ws) {
    int idx = blockIdx.x * 256 + threadIdx.x;
    if (idx < 512 + 1 + 3) ws[WS_COLSUM + idx] = 0.0f;
}

// ---- per-row squared norms: one 128-thread block per row -------------------
__global__ void k_rowsq(const float* __restrict__ src, const float* __restrict__ tgt,
                        float* __restrict__ sq) {
    int row = blockIdx.x;
    const float* p = row_ptr(src, tgt, row);
    int t = threadIdx.x;                       // 128 threads * 4 floats = 512
    float4 v = ((const float4*)p)[t];
    float s = v.x * v.x + v.y * v.y + v.z * v.z + v.w * v.w;
    __shared__ float red[128];
    red[t] = s;
    __syncthreads();
    for (int off = 64; off > 0; off >>= 1) {
        if (t < off) red[t] += red[t + off];
        __syncthreads();
    }
    if (t == 0) sq[row] = red[0];
}

// ---- column sums: 64 blocks x 256 threads; block handles 128 rows ----------
__global__ void k_colsum(const float* __restrict__ src, const float* __restrict__ tgt,
                         float* __restrict__ colsum) {
    int d0 = threadIdx.x;                      // dims d0 and d0+256
    int r0 = blockIdx.x * 128;
    float s0 = 0.0f, s1 = 0.0f;
    for (int r = 0; r < 128; ++r) {
        const float* p = row_ptr(src, tgt, r0 + r);
        s0 += p[d0];
        s1 += p[d0 + 256];
    }
    atomicAdd(&colsum[d0], s0);
    atomicAdd(&colsum[d0 + 256], s1);
}

// ---- bandwidth = (2N*sum(sq) - 2*||colsum||^2) / (N^2 - N) -----------------
__global__ void k_bw(const float* __restrict__ sq, const float* __restrict__ colsum,
                     float* __restrict__ bwout) {
    int t = threadIdx.x;
    float s = 0.0f;
    for (int i = t; i < NTOT; i += 256) s += sq[i];
    float cs = 0.0f;
    for (int i = t; i < D; i += 256) { float v = colsum[i]; cs += v * v; }
    __shared__ float r1[256], r2[256];
    r1[t] = s; r2[t] = cs;
    __syncthreads();
    for (int off = 128; off > 0; off >>= 1) {
        if (t < off) { r1[t] += r1[t + off]; r2[t] += r2[t + off]; }
        __syncthreads();
    }
    if (t == 0) {
        float suml2 = 2.0f * (float)NTOT * r1[0] - 2.0f * r2[0];
        bwout[0] = suml2 / ((float)NTOT * (float)NTOT - (float)NTOT);
    }
}

// ---- fused Gram-tile + RBF kernel-sum + region accumulation ----------------
// One wave computes a 64x64 macro tile of the Gram matrix via 4x4
// V_WMMA_F32_16X16X4_F32 tiles.  Strip base pointers are wave-uniform
// (readfirstlane -> SGPR pairs); one shared per-lane 32-bit offset selects
// the lane's row and K-pair, so loads use SADDR+VOFF addressing.
__global__ void __launch_bounds__(32 * WAVES_PER_BLOCK, 1)
k_gram(const float* __restrict__ src, const float* __restrict__ tgt,
       const float* __restrict__ sq, const float* __restrict__ bwp,
       float* __restrict__ accum) {
    const int wave = threadIdx.x >> 5;
    const int lane = threadIdx.x & 31;
    // tile id is identical across the wave: force it scalar
    const int tid  = __builtin_amdgcn_readfirstlane(blockIdx.x * WAVES_PER_BLOCK + wave);
    const int ti   = tid >> 7;                 // / TPD (=128)
    const int tj   = tid & (TPD - 1);
    const int iBase = ti * TS;                 // scalar
    const int jBase = tj * TS;                 // scalar
    const int lm = lane & 15;                  // M/N index within 16
    const int h  = lane >> 4;                  // half-wave selector

    // 5 bandwidth constants; fold log2(e) into them: exp(-l2/b) = exp2(-l2*c)
    const float bw = bwp[0];
    const float mul[5] = {0.25f, 0.5f, 1.0f, 2.0f, 4.0f};
    float c[5];
#pragma unroll
    for (int b = 0; b < 5; ++b)
        c[b] = 1.442695040888963f / (bw * mul[b] + 1e-8f);

    // Scalar strip base pointers (each 16-row strip entirely in src or tgt
    // because NS is a multiple of TS).
    const float* baseA[4];
    const float* baseB[4];
#pragma unroll
    for (int s = 0; s < 4; ++s) {
        int ra = iBase + 16 * s;
        int rb = jBase + 16 * s;
        baseA[s] = (ra < NS) ? (src + (size_t)ra * D) : (tgt + (size_t)(ra - NS) * D);
        baseB[s] = (rb < NS) ? (src + (size_t)rb * D) : (tgt + (size_t)(rb - NS) * D);
    }
    // Single per-lane element offset, shared by all 8 strips:
    // lane reads row (lane&15) of the strip at columns k+2h, k+2h+1.
    const int voff = lm * D + 2 * h;

    v8f acc[4][4];
#pragma unroll
    for (int s = 0; s < 4; ++s)
#pragma unroll
        for (int t = 0; t < 4; ++t) acc[s][t] = (v8f)(0.0f);

    for (int k = 0; k < D; k += 4) {
        v2f a[4], b[4];
#pragma unroll
        for (int s = 0; s < 4; ++s) a[s] = *(const v2f*)(baseA[s] + voff + k);
#pragma unroll
        for (int t = 0; t < 4; ++t) b[t] = *(const v2f*)(baseB[t] + voff + k);
#pragma unroll
        for (int s = 0; s < 4; ++s)
#pragma unroll
            for (int t = 0; t < 4; ++t)
                acc[s][t] = __builtin_amdgcn_wmma_f32_16x16x4_f32(
                    false, a[s], false, b[t], (short)0, acc[s][t], false, false);
    }

    // Epilogue: C/D layout -> lane<16: M=r, N=lane; lane>=16: M=r+8, N=lane-16
    float local = 0.0f;
    float sqn[4];
#pragma unroll
    for (int t = 0; t < 4; ++t) sqn[t] = sq[jBase + 16 * t + lm];
#pragma unroll
    for (int s = 0; s < 4; ++s) {
        float sqa[8];
#pragma unroll
        for (int r = 0; r < 8; ++r) sqa[r] = sq[iBase + 16 * s + 8 * h + r];
#pragma unroll
        for (int t = 0; t < 4; ++t) {
#pragma unroll
            for (int r = 0; r < 8; ++r) {
                float g  = acc[s][t][r];
                float l2 = fmaxf(sqa[r] + sqn[t] - 2.0f * g, 0.0f);
#pragma unroll
                for (int b = 0; b < 5; ++b)
                    local += __builtin_amdgcn_exp2f(-l2 * c[b]);
            }
        }
    }

    // wave32 reduction
#pragma unroll
    for (int off = 16; off > 0; off >>= 1)
        local += __shfl_xor(local, off, 32);

    if (lane == 0) {
        int region;
        if (iBase < NS && jBase < NS)        region = 0;  // xx
        else if (iBase >= NS && jBase >= NS) region = 1;  // yy
        else                                 region = 2;  // xy (both blocks)
        atomicAdd(&accum[region], local);
    }
}

// ---- final combine ---------------------------------------------------------
// xx = Sxx/ns^2; yy = Syy/ns^2; xy = Sxy/(2*ns^2)  (Sxy holds both off blocks)
// result = xx + yy - 2*xy = (Sxx + Syy - Sxy) / ns^2
__global__ void k_final(const float* __restrict__ accum, float* __restrict__ out) {
    out[0] = (accum[0] + accum[1] - accum[2]) / ((float)NS * (float)NS);
}

extern "C" void kernel_launch(void* const* d_in, const int* in_sizes, int n_in,
                              void* d_out, int out_size, void* d_ws, size_t ws_size,
                              hipStream_t stream) {
    (void)in_sizes; (void)n_in; (void)out_size; (void)ws_size;
    const float* src = (const float*)d_in[0];
    const float* tgt = (const float*)d_in[1];
    float* ws     = (float*)d_ws;
    float* sq     = ws + WS_SQ;
    float* colsum = ws + WS_COLSUM;
    float* bw     = ws + WS_BW;
    float* accum  = ws + WS_ACC;
    float* out    = (float*)d_out;

    k_init  <<<3,    256, 0, stream>>>(ws);
    k_rowsq <<<NTOT, 128, 0, stream>>>(src, tgt, sq);
    k_colsum<<<64,   256, 0, stream>>>(src, tgt, colsum);
    k_bw    <<<1,    256, 0, stream>>>(sq, colsum, bw);
    k_gram  <<<NTILES / WAVES_PER_BLOCK, 32 * WAVES_PER_BLOCK, 0, stream>>>(
        src, tgt, sq, bw, accum);
    k_final <<<1, 1, 0, stream>>>(accum, out);
}